// MSRPLoss_57389353009458
// MI455X (gfx1250) — compile-verified
//
#include <hip/hip_runtime.h>

typedef float v2f __attribute__((ext_vector_type(2)));
typedef float v4f __attribute__((ext_vector_type(4)));
typedef float v8f __attribute__((ext_vector_type(8)));

// Seed the output with -(N): accounts for the excluded j==label terms
// (each contributes exactly relu(1)=1 to the unmasked sum).
__global__ void msrp_init_out(float* out, float init) { out[0] = init; }

__global__ __launch_bounds__(256) void msrp_hinge_sum_wmma(
    const float* __restrict__ outs,
    const long long* __restrict__ labels,
    float* __restrict__ result,
    int N, int C)
{
    const int lane        = threadIdx.x & 31;
    const int wave_in_blk = threadIdx.x >> 5;
    // Broadcast wave ids into SGPRs so both loops have scalar trip counts and
    // EXEC stays all-1s (required for WMMA, and kills per-iter mask ops).
    const int gwave  = __builtin_amdgcn_readfirstlane(blockIdx.x * 8 + wave_in_blk);
    const int nwaves = __builtin_amdgcn_readfirstlane(gridDim.x * 8);

    const v2f ones = {1.0f, 1.0f};
    v8f acc = {};                                  // f32 16x16 accumulator (8 VGPRs)

    const int vecC  = C >> 2;                      // float4s per row (1024)
    const int iters = vecC >> 5;                   // per-lane float4s per row (32)

    // One row per wave: bias is wave-uniform, loads are lane-contiguous b128.
    for (int row = gwave; row < N; row += nwaves) {
        const size_t base = (size_t)row * (size_t)C;
        const float  g    = outs[base + (size_t)labels[row]];
        const float  bias = 1.0f - g;
        const v4f* __restrict__ p = (const v4f*)(outs + base);

#pragma unroll 2
        for (int jj = 0; jj < iters; ++jj) {
            v4f v = __builtin_nontemporal_load(p + ((jj << 5) + lane));
            v2f a0 = { fmaxf(v.x + bias, 0.0f), fmaxf(v.y + bias, 0.0f) };
            v2f a1 = { fmaxf(v.z + bias, 0.0f), fmaxf(v.w + bias, 0.0f) };
            // D = A(16x4) * ones(4x16) + C : folds 64 f32 margins into acc per op.
            acc = __builtin_amdgcn_wmma_f32_16x16x4_f32(
                false, a0, false, ones, (short)0, acc, false, false);
            acc = __builtin_amdgcn_wmma_f32_16x16x4_f32(
                false, a1, false, ones, (short)0, acc, false, false);
        }
    }

    // Each of the 16 D columns holds the identical row-sum vector, so the sum
    // of all 256 accumulator entries across the wave is 16x the true total.
    float s = acc[0] + acc[1] + acc[2] + acc[3] + acc[4] + acc[5] + acc[6] + acc[7];
    for (int off = 16; off > 0; off >>= 1)
        s += __shfl_down(s, off, 32);

    __shared__ float wsum[8];
    if (lane == 0) wsum[wave_in_blk] = s;
    __syncthreads();
    if (threadIdx.x == 0) {
        float t = 0.0f;
        for (int i = 0; i < 8; ++i) t += wsum[i];
        atomicAdd(result, t * 0.0625f);            // divide out the 16x replication
    }
}

extern "C" void kernel_launch(void* const* d_in, const int* in_sizes, int n_in,
                              void* d_out, int out_size, void* d_ws, size_t ws_size,
                              hipStream_t stream) {
    const float*     outs   = (const float*)d_in[0];
    const long long* labels = (const long long*)d_in[1];
    float*           out    = (float*)d_out;

    const int N = in_sizes[1];                  // 16384
    const int C = in_sizes[0] / N;              // 4096

    msrp_init_out<<<1, 1, 0, stream>>>(out, -(float)N);
    msrp_hinge_sum_wmma<<<512, 256, 0, stream>>>(outs, labels, out, N, C);
}